// Condition_LSTM2_5772436046453
// MI455X (gfx1250) — compile-verified
//
#include <hip/hip_runtime.h>
#include <hip/hip_bf16.h>
#include <math.h>

// Problem constants
#define BB   2048
#define II   3
#define AA   80
#define HH   1024
#define KW   10
#define UU   64
#define OUTN 121        // 6*KC+1
#define K1P  84         // (I+A)=83 padded to 84 (mult of 4)
#define K2P  1108       // (I+A+H)=1107 padded to 1108

typedef float v2f __attribute__((ext_vector_type(2)));
typedef float v8f __attribute__((ext_vector_type(8)));

__device__ __forceinline__ float sigf(float x) {
    return 1.0f / (1.0f + __expf(-x));
}
__device__ __forceinline__ float tanh_fast(float x) {
    float xc = fminf(fmaxf(x, -15.0f), 15.0f);
    float e2 = __expf(2.0f * xc);
    return (e2 - 1.0f) / (e2 + 1.0f);
}

// ---------------------------------------------------------------------------
// Packing kernels (build zero-padded concat inputs / weights in workspace)
// ---------------------------------------------------------------------------
__global__ void pack_in1(const float* __restrict__ x,
                         const float* __restrict__ w_t_pre,
                         float* __restrict__ in1p) {
    int b = blockIdx.x;
    int t = threadIdx.x;
    if (t < K1P) {
        float v = 0.0f;
        if (t < II)              v = x[b * II + t];
        else if (t < II + AA)    v = w_t_pre[b * AA + (t - II)];
        in1p[b * K1P + t] = v;
    }
}

__global__ void pack_in2(const float* __restrict__ x,
                         const float* __restrict__ w_t,
                         const float* __restrict__ h1,
                         float* __restrict__ in2p) {
    int b = blockIdx.x;
    for (int t = threadIdx.x; t < K2P; t += blockDim.x) {
        float v = 0.0f;
        if (t < II)                   v = x[b * II + t];
        else if (t < II + AA)         v = w_t[b * AA + (t - II)];
        else if (t < II + AA + HH)    v = h1[b * HH + (t - II - AA)];
        in2p[b * K2P + t] = v;
    }
}

// generic row-pad copy: src [R,Ks] -> dst [R,Kd], zero pad Ks..Kd-1
__global__ void pack_w(const float* __restrict__ src, int Ks,
                       float* __restrict__ dst, int Kd) {
    int r = blockIdx.x;
    for (int k = threadIdx.x; k < Kd; k += blockDim.x) {
        dst[(size_t)r * Kd + k] = (k < Ks) ? src[(size_t)r * Ks + k] : 0.0f;
    }
}

// ---------------------------------------------------------------------------
// Fused LSTM-gate GEMM:  gates[b, g*H+n] = A0[b,:]*B0[g*H+n,:] + A1[b,:]*B1[g*H+n,:]
// then LSTM nonlinearity in the epilogue.
// Block = 128 threads (4 waves). Tile = 128 M x 32 N x 4 gates.
// Each wave handles 32 M rows (2 subtiles) x 2 N-subtiles x 4 gates
// -> 16 accumulators; per K-step: 2 A-frag + 8 B-frag loads feed 16 wmma.
// ---------------------------------------------------------------------------
__device__ __forceinline__ void gemm_phase(const float* __restrict__ Ar0,
                                           const float* __restrict__ Ar1,
                                           const float* const* __restrict__ Br,
                                           int K, v8f acc[16]) {
    for (int k = 0; k < K; k += 4) {
        v2f a0 = *(const v2f*)(Ar0 + k);
        v2f a1 = *(const v2f*)(Ar1 + k);
#pragma unroll
        for (int i = 0; i < 8; ++i) {
            v2f bf = *(const v2f*)(Br[i] + k);
            acc[i * 2 + 0] = __builtin_amdgcn_wmma_f32_16x16x4_f32(
                false, a0, false, bf, (short)0, acc[i * 2 + 0], false, false);
            acc[i * 2 + 1] = __builtin_amdgcn_wmma_f32_16x16x4_f32(
                false, a1, false, bf, (short)0, acc[i * 2 + 1], false, false);
        }
    }
}

__global__ __launch_bounds__(128) void lstm_gemm(
    const float* __restrict__ A0, int sA0,            // packed input  [B, sA0]
    const float* __restrict__ B0, int sB0, int K0,    // packed W_ih   [4H, sB0]
    const float* __restrict__ A1,                     // h_prev [B, H]
    const float* __restrict__ B1,                     // W_hh   [4H, H]
    const float* __restrict__ b_ih, const float* __restrict__ b_hh,
    const float* __restrict__ c_prev,
    float* __restrict__ h_out, float* __restrict__ c_out) {

    const int tid  = threadIdx.x;
    const int lane = tid & 31;
    const int wv   = tid >> 5;                 // 0..3
    const int r    = lane & 15;
    const int hi   = lane >> 4;                // 0/1
    const int kh   = hi << 1;                  // K offset 0 or 2 (ISA A/B frag layout)
    const int m0   = blockIdx.x * 128;
    const int n0   = blockIdx.y * 32;
    const int mw   = m0 + wv * 32;             // wave's 32-row M range
    const int ma   = mw + r;                   // A-frag row, M-subtile 0
    const int mb   = mw + 16 + r;              // A-frag row, M-subtile 1

    v8f acc[16];
    const v8f vzero = {0.f, 0.f, 0.f, 0.f, 0.f, 0.f, 0.f, 0.f};
#pragma unroll
    for (int i = 0; i < 16; ++i) acc[i] = vzero;

    // Phase 0: packed input part
    {
        const float* Ar0 = A0 + (size_t)ma * sA0 + kh;
        const float* Ar1 = A0 + (size_t)mb * sA0 + kh;
        const float* Br[8];
#pragma unroll
        for (int g = 0; g < 4; ++g)
#pragma unroll
            for (int t = 0; t < 2; ++t)
                Br[g * 2 + t] = B0 + (size_t)(g * HH + n0 + t * 16 + r) * sB0 + kh;
        gemm_phase(Ar0, Ar1, Br, K0, acc);
    }
    // Phase 1: hidden-state part (K = H = 1024)
    {
        const float* Ar0 = A1 + (size_t)ma * HH + kh;
        const float* Ar1 = A1 + (size_t)mb * HH + kh;
        const float* Br[8];
#pragma unroll
        for (int g = 0; g < 4; ++g)
#pragma unroll
            for (int t = 0; t < 2; ++t)
                Br[g * 2 + t] = B1 + (size_t)(g * HH + n0 + t * 16 + r) * HH + kh;
        gemm_phase(Ar0, Ar1, Br, HH, acc);
    }

    // Epilogue: C/D layout — VGPR p: lanes 0-15 -> M=p, lanes 16-31 -> M=p+8; N = lane&15
#pragma unroll
    for (int t = 0; t < 2; ++t) {
        const int n  = n0 + t * 16 + r;
        const float bi = b_ih[n]           + b_hh[n];
        const float bf = b_ih[HH + n]      + b_hh[HH + n];
        const float bg = b_ih[2 * HH + n]  + b_hh[2 * HH + n];
        const float bo = b_ih[3 * HH + n]  + b_hh[3 * HH + n];
#pragma unroll
        for (int u = 0; u < 2; ++u) {
            const int mbase = mw + u * 16 + (hi << 3);
#pragma unroll
            for (int p = 0; p < 8; ++p) {
                const int mm = mbase + p;
                float iv = acc[(0 * 2 + t) * 2 + u][p] + bi;
                float fv = acc[(1 * 2 + t) * 2 + u][p] + bf;
                float gv = acc[(2 * 2 + t) * 2 + u][p] + bg;
                float ov = acc[(3 * 2 + t) * 2 + u][p] + bo;
                float cp = c_prev[(size_t)mm * HH + n];
                float cn = sigf(fv) * cp + sigf(iv) * tanh_fast(gv);
                float hn = sigf(ov) * tanh_fast(cn);
                c_out[(size_t)mm * HH + n] = cn;
                h_out[(size_t)mm * HH + n] = hn;
            }
        }
    }
}

// ---------------------------------------------------------------------------
// Gaussian-mixture attention window: one block per batch row.
// ---------------------------------------------------------------------------
__global__ __launch_bounds__(256) void attention_kernel(
    const float* __restrict__ h1, const float* __restrict__ W_abk,
    const float* __restrict__ b_abk, const float* __restrict__ kappa_prev,
    const float* __restrict__ c_seq,
    float* __restrict__ kappa_out, float* __restrict__ wt_out) {

    __shared__ float sh[HH];
    __shared__ float sred[256];
    __shared__ float s_abk[32];
    __shared__ float s_alpha[KW], s_beta[KW], s_kap[KW];
    __shared__ float s_phi[UU];

    const int b   = blockIdx.x;
    const int tid = threadIdx.x;

    for (int i = tid; i < HH; i += 256) sh[i] = h1[(size_t)b * HH + i];
    __syncthreads();

    // abk = h1 @ W_abk^T  (30 outputs, split K over 8 partials)
    const int j = tid & 31, part = tid >> 5;
    float s = 0.0f;
    if (j < 3 * KW) {
        const float* w  = W_abk + (size_t)j * HH + part * 128;
        const float* hh = sh + part * 128;
        for (int k = 0; k < 128; ++k) s += hh[k] * w[k];
    }
    sred[tid] = s;
    __syncthreads();
    if (tid < 32) {
        float tot = 0.0f;
#pragma unroll
        for (int p = 0; p < 8; ++p) tot += sred[p * 32 + tid];
        if (tid < 3 * KW) s_abk[tid] = tot + b_abk[tid];
    }
    __syncthreads();

    if (tid < KW) {
        float alpha = __expf(s_abk[tid]);
        float beta  = __expf(s_abk[KW + tid]);
        float kap   = kappa_prev[b * KW + tid] + __expf(s_abk[2 * KW + tid]);
        s_alpha[tid] = alpha;
        s_beta[tid]  = beta;
        s_kap[tid]   = kap;
        kappa_out[b * KW + tid] = kap;
    }
    __syncthreads();

    if (tid < UU) {
        float u   = (float)tid;
        float phi = 0.0f;
#pragma unroll
        for (int w = 0; w < KW; ++w) {
            float d = s_kap[w] - u;
            phi += s_alpha[w] * __expf(-s_beta[w] * d * d);
        }
        s_phi[tid] = phi;
    }
    __syncthreads();

    if (tid < AA) {
        float acc = 0.0f;
        const float* cs = c_seq + (size_t)b * UU * AA + tid;
#pragma unroll 4
        for (int u = 0; u < UU; ++u) acc += s_phi[u] * cs[(size_t)u * AA];
        wt_out[b * AA + tid] = acc;
    }
}

// ---------------------------------------------------------------------------
// Output head: out = h2 @ W_fc2^T + b_fc2   (121 outputs per row)
// ---------------------------------------------------------------------------
__global__ __launch_bounds__(256) void fc2_kernel(
    const float* __restrict__ h2, const float* __restrict__ W_fc2,
    const float* __restrict__ b_fc2, float* __restrict__ out) {

    __shared__ float sh[HH];
    __shared__ float sred[256];
    const int b   = blockIdx.x;
    const int tid = threadIdx.x;

    for (int i = tid; i < HH; i += 256) sh[i] = h2[(size_t)b * HH + i];
    __syncthreads();

    const int o = tid & 127, part = tid >> 7;   // 2 partials of 512
    float s = 0.0f;
    if (o < OUTN) {
        const float* w  = W_fc2 + (size_t)o * HH + part * 512;
        const float* hh = sh + part * 512;
        for (int k = 0; k < 512; ++k) s += hh[k] * w[k];
    }
    sred[tid] = s;
    __syncthreads();
    if (tid < OUTN) {
        out[(size_t)b * OUTN + tid] = sred[tid] + sred[128 + tid] + b_fc2[tid];
    }
}

// ---------------------------------------------------------------------------
extern "C" void kernel_launch(void* const* d_in, const int* in_sizes, int n_in,
                              void* d_out, int out_size, void* d_ws, size_t ws_size,
                              hipStream_t stream) {
    (void)in_sizes; (void)n_in; (void)out_size; (void)ws_size;

    const float* x          = (const float*)d_in[0];
    const float* c_seq      = (const float*)d_in[1];
    const float* h1_prev    = (const float*)d_in[2];
    const float* c1_prev    = (const float*)d_in[3];
    const float* h2_prev    = (const float*)d_in[4];
    const float* c2_prev    = (const float*)d_in[5];
    const float* kappa_prev = (const float*)d_in[6];
    const float* w_t_pre    = (const float*)d_in[7];
    const float* W_ih1      = (const float*)d_in[8];
    const float* W_hh1      = (const float*)d_in[9];
    const float* b_ih1      = (const float*)d_in[10];
    const float* b_hh1      = (const float*)d_in[11];
    const float* W_ih2      = (const float*)d_in[12];
    const float* W_hh2      = (const float*)d_in[13];
    const float* b_ih2      = (const float*)d_in[14];
    const float* b_hh2      = (const float*)d_in[15];
    const float* W_abk      = (const float*)d_in[16];
    const float* b_abk      = (const float*)d_in[17];
    const float* W_fc2      = (const float*)d_in[18];
    const float* b_fc2      = (const float*)d_in[19];

    // workspace layout (floats)
    float* ws   = (float*)d_ws;
    float* in1p = ws;                               // [B, 84]
    float* w1p  = in1p + (size_t)BB * K1P;          // [4H, 84]
    float* in2p = w1p  + (size_t)4 * HH * K1P;      // [B, 1108]
    float* w2p  = in2p + (size_t)BB * K2P;          // [4H, 1108]

    // output layout (floats): out, h1, c1, h2, c2, kappa, w_t
    float* out = (float*)d_out;
    float* h1  = out + (size_t)BB * OUTN;
    float* c1  = h1  + (size_t)BB * HH;
    float* h2  = c1  + (size_t)BB * HH;
    float* c2  = h2  + (size_t)BB * HH;
    float* kap = c2  + (size_t)BB * HH;
    float* wt  = kap + (size_t)BB * KW;

    // LSTM1
    pack_in1<<<BB, 128, 0, stream>>>(x, w_t_pre, in1p);
    pack_w<<<4 * HH, 256, 0, stream>>>(W_ih1, II + AA, w1p, K1P);
    lstm_gemm<<<dim3(BB / 128, HH / 32), 128, 0, stream>>>(
        in1p, K1P, w1p, K1P, K1P, h1_prev, W_hh1, b_ih1, b_hh1, c1_prev, h1, c1);

    // attention window
    attention_kernel<<<BB, 256, 0, stream>>>(h1, W_abk, b_abk, kappa_prev,
                                             c_seq, kap, wt);

    // LSTM2
    pack_in2<<<BB, 256, 0, stream>>>(x, wt, h1, in2p);
    pack_w<<<4 * HH, 256, 0, stream>>>(W_ih2, II + AA + HH, w2p, K2P);
    lstm_gemm<<<dim3(BB / 128, HH / 32), 128, 0, stream>>>(
        in2p, K2P, w2p, K2P, K2P, h2_prev, W_hh2, b_ih2, b_hh2, c2_prev, h2, c2);

    // output head
    fc2_kernel<<<BB, 256, 0, stream>>>(h2, W_fc2, b_fc2, out);
}